// SAGE_29257317220563
// MI455X (gfx1250) — compile-verified
//
#include <hip/hip_runtime.h>

typedef __attribute__((ext_vector_type(2))) float v2f;
typedef __attribute__((ext_vector_type(8))) float v8f;

// ---------------- utility kernels ----------------

__global__ void zero_f32(float* __restrict__ p, long n) {
  long i = (long)blockIdx.x * blockDim.x + threadIdx.x;
  long stride = (long)gridDim.x * blockDim.x;
  for (; i < n; i += stride) p[i] = 0.0f;
}

__global__ void degree_k(const int* __restrict__ dst, float* __restrict__ deg, int E) {
  int e = blockIdx.x * blockDim.x + threadIdx.x;
  if (e < E) atomicAdd(&deg[dst[e]], 1.0f);
}

// 16 lanes per edge, 4 floats per lane: float4 gather from L2-resident t,
// 4x fp32 atomic adds into L2-resident agg.
__global__ void scatter_k(const float* __restrict__ t, const int* __restrict__ src,
                          const int* __restrict__ dst, float* __restrict__ agg, int E) {
  int idx = blockIdx.x * blockDim.x + threadIdx.x;
  int e = idx >> 4;
  if (e >= E) return;
  int l = idx & 15;
  int s = src[e], d = dst[e];
  const float4 v = *reinterpret_cast<const float4*>(t + (size_t)s * 64 + l * 4);
  float* p = agg + (size_t)d * 64 + l * 4;
  atomicAdd(p + 0, v.x);
  atomicAdd(p + 1, v.y);
  atomicAdd(p + 2, v.z);
  atomicAdd(p + 3, v.w);
}

// ---------------- WMMA GEMM ----------------
// One wave -> 16 rows x 64 cols fp32 tile. 8 waves/block -> 128 rows/block.
// K stepped by 4 using v_wmma_f32_16x16x4_f32.
// W staged in LDS *pair-interleaved*: Wl[(k/2)*128 + col*2 + (k%2)], so a
// B fragment (K-pair per lane) is one aligned ds_load_b64 into a VGPR pair.
// COMBINE: out = relu(acc + bias + agg/deg), else out = acc.
template <int K, bool COMBINE>
__global__ __launch_bounds__(256) void sage_gemm(
    const float* __restrict__ Hin, const float* __restrict__ W,
    const float* __restrict__ bias, const float* __restrict__ agg,
    const float* __restrict__ deg, float* __restrict__ Hout, int n) {
  __shared__ float Wl[K * 64];
  for (int i = threadIdx.x; i < K * 64; i += 256) {
    int k = i >> 6;
    int col = i & 63;
    Wl[(k >> 1) * 128 + col * 2 + (k & 1)] = W[i];
  }
  __syncthreads();

  const int wave = threadIdx.x >> 5;
  const int lane = threadIdx.x & 31;
  const int m    = lane & 15;     // row within tile for A / col for B
  const int half = lane >> 4;     // selects K-pair per ISA 32-bit A/B layout
  const int kk   = half * 2;

  const int row_base = blockIdx.x * 128 + wave * 16;
  int lrow = row_base + m;
  if (lrow >= n) lrow = n - 1;    // clamp: WMMA requires EXEC all-1s
  const float* __restrict__ Arow = Hin + (size_t)lrow * K;

  v8f acc[4] = {};

  for (int k0 = 0; k0 < K; k0 += 4) {
    // A fragment: lane m = M, VGPR r holds K = k0 + half*2 + r  (8B aligned)
    v2f a = *reinterpret_cast<const v2f*>(Arow + k0 + kk);
    // B fragments: interleaved LDS -> one b64 load per n-tile
    const float* bp = &Wl[((k0 + kk) >> 1) * 128 + m * 2];
#pragma unroll
    for (int t = 0; t < 4; ++t) {
      v2f b = *reinterpret_cast<const v2f*>(bp + t * 32);
      acc[t] = __builtin_amdgcn_wmma_f32_16x16x4_f32(
          /*neg_a=*/false, a, /*neg_b=*/false, b,
          /*c_mod=*/(short)0, acc[t], /*reuse_a=*/false, /*reuse_b=*/false);
    }
  }

  // Epilogue. C/D layout: VGPR v holds row M = v + 8*half, col N = m + 16*t.
  if (COMBINE) {
    float invd[8];
#pragma unroll
    for (int v = 0; v < 8; ++v) {
      int row = row_base + v + 8 * half;
      float d = (row < n) ? deg[row] : 1.0f;
      invd[v] = 1.0f / fmaxf(d, 1.0f);
    }
#pragma unroll
    for (int t = 0; t < 4; ++t) {
      int col = t * 16 + m;
      float bv = bias[col];
#pragma unroll
      for (int v = 0; v < 8; ++v) {
        int row = row_base + v + 8 * half;
        if (row < n) {
          float x = acc[t][v] + bv + agg[(size_t)row * 64 + col] * invd[v];
          Hout[(size_t)row * 64 + col] = fmaxf(x, 0.0f);
        }
      }
    }
  } else {
#pragma unroll
    for (int t = 0; t < 4; ++t) {
      int col = t * 16 + m;
#pragma unroll
      for (int v = 0; v < 8; ++v) {
        int row = row_base + v + 8 * half;
        if (row < n) Hout[(size_t)row * 64 + col] = acc[t][v];
      }
    }
  }
}

// ---------------- driver ----------------

extern "C" void kernel_launch(void* const* d_in, const int* in_sizes, int n_in,
                              void* d_out, int out_size, void* d_ws, size_t ws_size,
                              hipStream_t stream) {
  const float* x       = (const float*)d_in[0];
  const int*   src     = (const int*)d_in[1];
  const int*   dst     = (const int*)d_in[2];
  const float* Wself1  = (const float*)d_in[3];
  const float* Wneigh1 = (const float*)d_in[4];
  const float* b1      = (const float*)d_in[5];
  const float* Wself2  = (const float*)d_in[6];
  const float* Wneigh2 = (const float*)d_in[7];
  const float* b2      = (const float*)d_in[8];
  const float* Wself3  = (const float*)d_in[9];
  const float* Wneigh3 = (const float*)d_in[10];
  const float* b3      = (const float*)d_in[11];

  const int N = in_sizes[0] / 128;  // 100000
  const int E = in_sizes[1];        // 1600000

  float* ws  = (float*)d_ws;
  float* deg = ws;                        // N
  float* t   = deg + N;                   // N*64
  float* agg = t   + (size_t)N * 64;      // N*64
  float* h1  = agg + (size_t)N * 64;      // N*64
  float* h2  = h1  + (size_t)N * 64;      // N*64
  float* out = (float*)d_out;             // N*64

  dim3 blk(256);
  const int gemm_grid   = (N + 127) / 128;
  const int edge_grid   = (E + 255) / 256;
  const int edge_grid16 = (int)(((long)E * 16 + 255) / 256);
  const long aggN = (long)N * 64;

  // degrees (shared by all layers)
  zero_f32<<<1024, blk, 0, stream>>>(deg, N);
  degree_k<<<edge_grid, blk, 0, stream>>>(dst, deg, E);

  // ----- layer 1 (K=128): transform-then-aggregate -----
  sage_gemm<128, false><<<gemm_grid, blk, 0, stream>>>(x, Wneigh1, nullptr, nullptr, nullptr, t, N);
  zero_f32<<<2048, blk, 0, stream>>>(agg, aggN);
  scatter_k<<<edge_grid16, blk, 0, stream>>>(t, src, dst, agg, E);
  sage_gemm<128, true><<<gemm_grid, blk, 0, stream>>>(x, Wself1, b1, agg, deg, h1, N);

  // ----- layer 2 (K=64) -----
  sage_gemm<64, false><<<gemm_grid, blk, 0, stream>>>(h1, Wneigh2, nullptr, nullptr, nullptr, t, N);
  zero_f32<<<2048, blk, 0, stream>>>(agg, aggN);
  scatter_k<<<edge_grid16, blk, 0, stream>>>(t, src, dst, agg, E);
  sage_gemm<64, true><<<gemm_grid, blk, 0, stream>>>(h1, Wself2, b2, agg, deg, h2, N);

  // ----- layer 3 (K=64) -----
  sage_gemm<64, false><<<gemm_grid, blk, 0, stream>>>(h2, Wneigh3, nullptr, nullptr, nullptr, t, N);
  zero_f32<<<2048, blk, 0, stream>>>(agg, aggN);
  scatter_k<<<edge_grid16, blk, 0, stream>>>(t, src, dst, agg, E);
  sage_gemm<64, true><<<gemm_grid, blk, 0, stream>>>(h2, Wself3, b3, agg, deg, out, N);
}